// VegasMap_81131932221876
// MI455X (gfx1250) — compile-verified
//
#include <hip/hip_runtime.h>

// VEGAS adaptive-map kernel for gfx1250 (MI455X).
// Memory-bound: ~132MB HBM traffic @ 23.3 TB/s -> ~5.7us floor.
// Tables staged into LDS as interleaved (grid,inc) float2 pairs via CDNA5
// async global->LDS copies (ASYNCcnt); gathers are ds_load_b64; jacobian
// product uses wave32 quad shuffles (DPP).

typedef float v4f __attribute__((ext_vector_type(4)));
typedef float v2f __attribute__((ext_vector_type(2)));

__device__ __forceinline__ void async_g2l_b32(unsigned lds_byte_addr,
                                              unsigned long long gbase,
                                              unsigned gbyte_off) {
  // GVS mode: mem_addr = SADDR(64b) + VADDR(32b offset); VDST = LDS byte addr.
  asm volatile("global_load_async_to_lds_b32 %0, %1, %2"
               :
               : "v"(lds_byte_addr), "v"(gbyte_off), "s"(gbase)
               : "memory");
}

__global__ __launch_bounds__(256) void vegas_map_kernel(
    const float* __restrict__ y,
    const float* __restrict__ grid,
    const float* __restrict__ inc,
    float* __restrict__ x_out,
    float* __restrict__ jac_out,
    int batch, int ninc)
{
  // LDS layout: pair[d][i] = (grid[d][i], inc[d][i]) for i in [0,ninc), then
  // edge[d] = grid[d][ninc] (right edge, used only when y >= 1.0).
  extern __shared__ v2f pair_tab[];            // [D*ninc] float2 + D floats
  const int D = 16;
  const int icount = D * ninc;                 // number of pairs

  const unsigned lds_base   = (unsigned)(unsigned long long)(void*)&pair_tab[0];
  const unsigned pair_bytes = 8u * (unsigned)icount;
  const unsigned long long gb = (unsigned long long)grid;
  const unsigned long long ib = (unsigned long long)inc;

  // Stage interleaved pairs: two async b32 copies per (d,i).
  for (int d = 0; d < D; ++d) {
    const unsigned gsrc_row = 4u * (unsigned)(d * (ninc + 1));
    const unsigned isrc_row = 4u * (unsigned)(d * ninc);
    const unsigned ldst_row = lds_base + 8u * (unsigned)(d * ninc);
    for (int i = threadIdx.x; i < ninc; i += blockDim.x) {
      async_g2l_b32(ldst_row + 8u * (unsigned)i,      gb, gsrc_row + 4u * (unsigned)i);
      async_g2l_b32(ldst_row + 8u * (unsigned)i + 4u, ib, isrc_row + 4u * (unsigned)i);
    }
  }
  // Stage right-edge column grid[d][ninc].
  if (threadIdx.x < (unsigned)D) {
    const int d = (int)threadIdx.x;
    async_g2l_b32(lds_base + pair_bytes + 4u * (unsigned)d,
                  gb, 4u * (unsigned)(d * (ninc + 1) + ninc));
  }

  asm volatile("s_wait_asynccnt 0" ::: "memory");
  __syncthreads();

  const v2f*   __restrict__ gh_tab = pair_tab;                    // [D][ninc]
  const float* __restrict__ edge   = (const float*)(pair_tab + icount); // [D]

  const float nincf = (float)ninc;
  const int quarters = batch * (D / 4);            // 4 dims per thread-iter
  const int stride   = (int)(gridDim.x * blockDim.x);
  const v4f* __restrict__ y4 = (const v4f*)y;
  v4f* __restrict__ x4 = (v4f*)x_out;

#pragma unroll 2
  for (int q = (int)(blockIdx.x * blockDim.x + threadIdx.x); q < quarters; q += stride) {
    v4f yv = __builtin_nontemporal_load(&y4[q]);
    const int d0 = (q & 3) * 4;
    const float yc[4] = {yv.x, yv.y, yv.z, yv.w};
    float xr[4];
    float f = 1.0f;
#pragma unroll
    for (int j = 0; j < 4; ++j) {
      const int d = d0 + j;
      const float t  = yc[j] * nincf;
      const float fl = floorf(t);
      const int   iy = (int)fl;
      const float dy = t - fl;
      const bool valid = iy < ninc;
      int iyc = iy < 0 ? 0 : iy;
      const int iyi = iyc > (ninc - 1) ? (ninc - 1) : iyc;
      const v2f gh = gh_tab[d * ninc + iyi];     // (grid[d][iy], inc[d][iy])
      xr[j] = valid ? fmaf(gh.y, dy, gh.x) : edge[d];
      // valid: h; invalid: clamped pair gives inc[d][ninc-1] -> matches ref
      f *= gh.y * nincf;
    }
    v4f xo;
    xo.x = xr[0]; xo.y = xr[1]; xo.z = xr[2]; xo.w = xr[3];
    __builtin_nontemporal_store(xo, &x4[q]);

    // product across the 4 quarter-lanes of this sample (groups of 4 lanes,
    // consecutive q per lane; quarters % 4 == 0 so groups never split).
    f *= __shfl_xor(f, 1, 32);
    f *= __shfl_xor(f, 2, 32);
    if ((q & 3) == 0)
      __builtin_nontemporal_store(f, &jac_out[q >> 2]);
  }
}

extern "C" void kernel_launch(void* const* d_in, const int* in_sizes, int n_in,
                              void* d_out, int out_size, void* d_ws, size_t ws_size,
                              hipStream_t stream) {
  const float* y    = (const float*)d_in[0];
  const float* grid = (const float*)d_in[1];
  const float* inc  = (const float*)d_in[2];
  // Derive shapes from flat sizes: grid=[D,ninc+1], inc=[D,ninc], y=[B,D]
  const int s_grid = in_sizes[1];
  const int s_inc  = in_sizes[2];
  int D    = s_grid - s_inc;          // (ninc+1)*D - ninc*D = D  (== 16)
  if (D <= 0) D = 16;
  const int ninc  = s_inc / D;        // 1000
  const int batch = in_sizes[0] / D;  // 1048576

  float* x_out   = (float*)d_out;
  float* jac_out = x_out + (size_t)batch * (size_t)D;

  const int threads  = 256;
  const long long quarters = (long long)batch * (D / 4);
  int blocks = (int)((quarters + threads - 1) / threads);
  if (blocks > 512) blocks = 512;     // 128KB LDS/wg -> 2 wg/WGP; ~512 resident
                                      // on a ~256-WGP part; caps redundant
                                      // L2->LDS table fill at 64MB total.

  // D*ninc float2 pairs + D edge floats = 128,064 B (fits 320KB/WGP, 2 per WGP)
  const size_t lds_bytes = (size_t)(D * ninc) * sizeof(float) * 2 + (size_t)D * sizeof(float);

  vegas_map_kernel<<<blocks, threads, lds_bytes, stream>>>(
      y, grid, inc, x_out, jac_out, batch, ninc);
}